// HausdorffDistance_12352325943647
// MI455X (gfx1250) — compile-verified
//
#include <hip/hip_runtime.h>
#include <math.h>

typedef float v2f __attribute__((ext_vector_type(2)));
typedef float v8f __attribute__((ext_vector_type(8)));

#define NPTS   4096
#define TILE   16
#define NTILES (NPTS / TILE)      // 256 column tiles
#define BLOCK  256                // 8 waves (wave32)
#define WPB    8                  // waves per block
#define NBT    8                  // B*T = 2*4

__global__ void hd_init_kernel(float* acc) {
    if (threadIdx.x < 2) acc[threadIdx.x] = 0.0f;
}

__global__ void hd_final_kernel(const float* acc, float* out) {
    if (threadIdx.x < 2) out[threadIdx.x] = acc[threadIdx.x] * (1.0f / 16384.0f); // /(T*N)
}

__global__ __launch_bounds__(BLOCK)
void hd_chamfer_wmma_kernel(const float* __restrict__ d1,
                            const float* __restrict__ d2,
                            float* __restrict__ acc) {
    // B tiles pre-packed in the exact WMMA B-operand layout, norm folded into K3:
    // tile mt, lane l -> 2 consecutive f32 at bpack[mt*64 + l*2].
    //   lanes 0-15  (half 0): (x, y)      of column point  (K0, K1)
    //   lanes 16-31 (half 1): (z, |b|^2)  of column point  (K2, K3)
    __shared__ __align__(16) float bpack[NTILES * 64];   // 64 KB

    const int block     = blockIdx.x;                     // 256 blocks
    const int bt        = block / (NTILES / WPB);         // 0..7, 32 blocks each
    const int tile_base = (block % (NTILES / WPB)) * WPB;

    const float* __restrict__ set1 = d1 + (size_t)bt * NPTS * 3;
    const float* __restrict__ set2 = d2 + (size_t)bt * NPTS * 3;

    // ---- Pack set2 into WMMA layout (one-time; 16 points per thread) ----
    for (int m = threadIdx.x; m < NPTS; m += BLOCK) {
        const float bx = set2[m * 3 + 0];
        const float by = set2[m * 3 + 1];
        const float bz = set2[m * 3 + 2];
        const int  t  = m >> 4;        // tile
        const int  c  = m & 15;        // column within tile
        float* base = &bpack[t * 64];
        base[c * 2 + 0]      = bx;                            // K0 = x
        base[c * 2 + 1]      = by;                            // K1 = y
        base[32 + c * 2 + 0] = bz;                            // K2 = z
        base[32 + c * 2 + 1] = bx * bx + by * by + bz * bz;   // K3 = |b|^2
    }
    __syncthreads();

    const int wave  = threadIdx.x >> 5;   // 0..7
    const int lane  = threadIdx.x & 31;
    const int half  = lane >> 4;
    const int idx16 = lane & 15;

    // ---- A tile: A = [-2x, -2y, -2z, 1], 16x4 fp32 ----
    // VGPR0 = K0 (lanes 0-15) / K2 (lanes 16-31); VGPR1 = K1 / K3.
    // |a|^2 is row-constant -> pulled out of the min, added after reduction.
    const int row_tile = tile_base + wave;            // 0..255
    const int r        = row_tile * TILE + idx16;     // point index in set1
    const float ax = set1[r * 3 + 0];
    const float ay = set1[r * 3 + 1];
    const float az = set1[r * 3 + 2];
    const float na = ax * ax + ay * ay + az * az;     // kept in a register

    v2f A;
    A.x = half ? (-2.0f * az) : (-2.0f * ax);
    A.y = half ? 1.0f         : (-2.0f * ay);

    v8f minv;
#pragma unroll
    for (int v = 0; v < 8; ++v) minv[v] = 3.402823466e38f;

    const v2f* __restrict__ bl = (const v2f*)&bpack[lane * 2];   // stride 32 v2f per tile

    // ---- Sweep all 256 column tiles: ds_load_b64 + wmma + min ----
    for (int mt = 0; mt < NTILES; ++mt) {
        const v2f B = bl[mt * 32];        // immediate-offset ds_load_b64

        v8f c = {};                       // SRC2 = inline 0
        // D = -2 a.b + |b|^2   (16x16 fp32 tile)
        c = __builtin_amdgcn_wmma_f32_16x16x4_f32(
                /*neg_a=*/false, A, /*neg_b=*/false, B,
                /*c_mod=*/(short)0, c, /*reuse_a=*/false, /*reuse_b=*/false);

#pragma unroll
        for (int v = 0; v < 8; ++v) minv[v] = fminf(minv[v], c[v]);
    }

    // ---- Min across the 16 column-lanes of each half (wave32) ----
#pragma unroll
    for (int v = 0; v < 8; ++v) {
        float m = minv[v];
        m = fminf(m, __shfl_xor(m, 1, 32));
        m = fminf(m, __shfl_xor(m, 2, 32));
        m = fminf(m, __shfl_xor(m, 4, 32));
        m = fminf(m, __shfl_xor(m, 8, 32));
        minv[v] = m;   // all lanes of half h hold mins for rows {8h..8h+7}
    }

    // ---- Per-row fixup: row j handled by lane 16*(j>>3) + j, which owns na_j ----
    const int sel = idx16 & 7;
    float m = minv[0];
#pragma unroll
    for (int v = 1; v < 8; ++v) m = (sel == v) ? minv[v] : m;

    float s = ((idx16 >> 3) == half) ? sqrtf(fmaxf(m + na, 0.0f)) : 0.0f;

    // full wave32 sum (each of the 16 rows contributed exactly once)
    s += __shfl_xor(s, 1, 32);
    s += __shfl_xor(s, 2, 32);
    s += __shfl_xor(s, 4, 32);
    s += __shfl_xor(s, 8, 32);
    s += __shfl_xor(s, 16, 32);

    if (lane == 0) {
        atomicAdd(&acc[bt >> 2], s);   // bt = b*4 + t -> batch b = bt/4
    }
}

extern "C" void kernel_launch(void* const* d_in, const int* in_sizes, int n_in,
                              void* d_out, int out_size, void* d_ws, size_t ws_size,
                              hipStream_t stream) {
    const float* d1 = (const float*)d_in[0];
    const float* d2 = (const float*)d_in[1];
    // d_in[2] is dim==0 scalar -> identity path, ignored.
    float* out = (float*)d_out;
    float* acc = (float*)d_ws;     // 2-float accumulator

    hd_init_kernel<<<1, 32, 0, stream>>>(acc);
    hd_chamfer_wmma_kernel<<<dim3(NBT * (NTILES / WPB)), dim3(BLOCK), 0, stream>>>(d1, d2, acc);
    hd_final_kernel<<<1, 32, 0, stream>>>(acc, out);
}